// AdaptGNN_78056735637936
// MI455X (gfx1250) — compile-verified
//
#include <hip/hip_runtime.h>

// AdaptGNN for MI455X (gfx1250, wave32).
// B=8, N=2048, D=H=128. Three layers of:
//   h = h@W + b ; nh = l2norm(h) ; out = (ew .* (nh nh^T)) @ h ; relu (not last)
// Big matmuls use V_WMMA_F32_16X16X4_F32; cos tiles never touch HBM.
// Tile staging uses CDNA5 async global->LDS loads (ASYNCcnt).

constexpr int Bb = 8;
constexpr int Nn = 2048;
constexpr int Dd = 128;
constexpr int LDSP = 132;   // padded LDS row stride (floats): kills bank conflicts

typedef __attribute__((ext_vector_type(2))) float v2f;
typedef __attribute__((ext_vector_type(8))) float v8f;

// ---------------------------------------------------------------------------
// Kernel A: h1 = hin @ W + b ; nh = h1 / max(||h1||, 1e-12). One block per row.
// ---------------------------------------------------------------------------
__global__ __launch_bounds__(128) void adaptgnn_linear_norm(
    const float* __restrict__ hin, const float* __restrict__ W,
    const float* __restrict__ bias, float* __restrict__ hout,
    float* __restrict__ nhout) {
  __shared__ float row[Dd];
  __shared__ float red[Dd];
  const size_t r = blockIdx.x;
  const int j = threadIdx.x;
  row[j] = hin[r * Dd + j];
  __syncthreads();
  float acc = bias[j];
#pragma unroll 8
  for (int k = 0; k < Dd; ++k) acc = fmaf(row[k], W[k * Dd + j], acc);
  hout[r * Dd + j] = acc;
  red[j] = acc * acc;
  __syncthreads();
  for (int s = 64; s > 0; s >>= 1) {
    if (j < s) red[j] += red[j + s];
    __syncthreads();
  }
  const float denom = fmaxf(sqrtf(red[0]), 1e-12f);
  nhout[r * Dd + j] = acc / denom;
}

// Async 64x128 f32 tile stage: global(row stride 128) -> LDS(row stride 132).
// Fully unrolled (16 straight-line b128 async issues per thread, no exec-mask
// loop). Tracked by ASYNCcnt.
__device__ __forceinline__ void adaptgnn_async_tile_load(
    const float* __restrict__ src, float* dst) {
  const int tid = threadIdx.x;
  const int r0 = tid >> 5;        // 0..3
  const int c4 = tid & 31;        // 16B chunk within row
#pragma unroll
  for (int ii = 0; ii < 16; ++ii) {
    const int row = ii * 4 + r0;
    const unsigned lds = (unsigned)(size_t)&dst[row * LDSP + c4 * 4];
    const float* g = src + (size_t)row * Dd + c4 * 4;
    asm volatile("global_load_async_to_lds_b128 %0, %1, off"
                 :: "v"(lds), "v"(g) : "memory");
  }
}

__device__ __forceinline__ void adaptgnn_wait_async() {
  asm volatile("s_wait_asynccnt 0x0" ::: "memory");
}

// ---------------------------------------------------------------------------
// Kernel B: out[p,:] = sum_q (ew[p,q] * (nh_p . nh_q)) * h[q,:]   (+ReLU)
// One workgroup = (batch, 64-row p-tile), 4 waves, each wave owns 16 rows.
// 256 WGs; ~101 KB LDS and <=341 VGPRs so three WGs co-reside per WGP.
// ---------------------------------------------------------------------------
__global__ __launch_bounds__(128, 3) void adaptgnn_fused_cos_agg(
    const float* __restrict__ h, const float* __restrict__ nh,
    const float* __restrict__ ew, float* __restrict__ out, const int relu) {
  __shared__ float sQ[Dd * LDSP];        // staged nh_q, then h_q (reused)
  __shared__ float sS[4 * 16 * LDSP];    // per-wave masked cos strips

  const int b    = blockIdx.x >> 5;      // 32 p-tiles per batch
  const int p0   = (blockIdx.x & 31) * 64;
  const int wave = threadIdx.x >> 5;     // 0..3
  const int lane = threadIdx.x & 31;
  const int m    = lane & 15;            // row (A/C) or col (B/C) index
  const int hi   = lane >> 4;            // half-wave select

  // A strip: nh rows [p0+wave*16 .. +15] in WMMA 16x4-per-step A layout.
  // A element (m,k): lane = m + 16*(k mod 4 >= 2), vgpr = k&1.
  const float* nhp = nh + ((size_t)b * Nn + p0 + wave * 16 + m) * Dd;
  float aS[64];
#pragma unroll
  for (int kb = 0; kb < 32; ++kb) {
    aS[kb * 2 + 0] = nhp[kb * 4 + hi * 2 + 0];
    aS[kb * 2 + 1] = nhp[kb * 4 + hi * 2 + 1];
  }

  const v8f vzero = {0.f, 0.f, 0.f, 0.f, 0.f, 0.f, 0.f, 0.f};
  v8f acc[8];
#pragma unroll
  for (int t = 0; t < 8; ++t) acc[t] = vzero;

  float* ss = &sS[wave * 16 * LDSP];
  const float* ewrow = ew + (size_t)b * Nn * Nn + (size_t)(p0 + wave * 16) * Nn;

  for (int q0 = 0; q0 < Nn; q0 += 128) {
    __syncthreads();
    adaptgnn_async_tile_load(nh + ((size_t)b * Nn + q0) * Dd, sQ);
    adaptgnn_wait_async();
    __syncthreads();

    // Stage 1: S = ew .* (nh_p @ nh_q^T), one 16x128 strip per wave -> LDS.
    // Two qt tiles at a time, K split in halves: 4 independent WMMA chains.
    for (int qt = 0; qt < 8; qt += 2) {
      v8f c0l = vzero, c0h = vzero, c1l = vzero, c1h = vzero;
#pragma unroll
      for (int kb = 0; kb < 16; ++kb) {
        const int kA = kb * 4 + hi * 2;        // K blocks 0..63
        const int kB = 64 + kb * 4 + hi * 2;   // K blocks 64..127
        v2f a0, a1, b0, b1, b2, b3;
        a0[0] = aS[kb * 2 + 0];
        a0[1] = aS[kb * 2 + 1];
        a1[0] = aS[(kb + 16) * 2 + 0];
        a1[1] = aS[(kb + 16) * 2 + 1];
        // B[k][n] = nh_q[n][k]; lanes 0..15 hit banks 4n+k (conflict-free)
        b0[0] = sQ[(qt * 16 + m) * LDSP + kA + 0];
        b0[1] = sQ[(qt * 16 + m) * LDSP + kA + 1];
        b1[0] = sQ[(qt * 16 + m) * LDSP + kB + 0];
        b1[1] = sQ[(qt * 16 + m) * LDSP + kB + 1];
        b2[0] = sQ[((qt + 1) * 16 + m) * LDSP + kA + 0];
        b2[1] = sQ[((qt + 1) * 16 + m) * LDSP + kA + 1];
        b3[0] = sQ[((qt + 1) * 16 + m) * LDSP + kB + 0];
        b3[1] = sQ[((qt + 1) * 16 + m) * LDSP + kB + 1];
        c0l = __builtin_amdgcn_wmma_f32_16x16x4_f32(false, a0, false, b0,
                                                    (short)0, c0l, false, false);
        c0h = __builtin_amdgcn_wmma_f32_16x16x4_f32(false, a1, false, b1,
                                                    (short)0, c0h, false, false);
        c1l = __builtin_amdgcn_wmma_f32_16x16x4_f32(false, a0, false, b2,
                                                    (short)0, c1l, false, false);
        c1h = __builtin_amdgcn_wmma_f32_16x16x4_f32(false, a1, false, b3,
                                                    (short)0, c1h, false, false);
      }
      const v8f c0 = c0l + c0h;
      const v8f c1 = c1l + c1h;
      // Mask with edge weights (direct global read) and park strips in LDS.
#pragma unroll
      for (int r = 0; r < 8; ++r) {
        const int row = r + 8 * hi;   // C/D layout: vgpr r, half-wave hi
        const float w0 = ewrow[(size_t)row * Nn + q0 + qt * 16 + m];
        const float w1 = ewrow[(size_t)row * Nn + q0 + (qt + 1) * 16 + m];
        ss[row * LDSP + qt * 16 + m] = c0[r] * w0;
        ss[row * LDSP + (qt + 1) * 16 + m] = c1[r] * w1;
      }
    }

    __syncthreads();                       // also orders ss writes vs reads
    adaptgnn_async_tile_load(h + ((size_t)b * Nn + q0) * Dd, sQ);
    adaptgnn_wait_async();
    __syncthreads();

    // Stage 2: acc += S_strip @ h_q (8 independent accumulator chains)
#pragma unroll
    for (int ht = 0; ht < 8; ++ht) {
      v8f c = acc[ht];
#pragma unroll
      for (int kb = 0; kb < 32; ++kb) {
        const int k0 = kb * 4 + hi * 2;
        v2f a, bb;
        a[0] = ss[m * LDSP + k0 + 0];
        a[1] = ss[m * LDSP + k0 + 1];
        bb[0] = sQ[(k0 + 0) * LDSP + ht * 16 + m];
        bb[1] = sQ[(k0 + 1) * LDSP + ht * 16 + m];
        c = __builtin_amdgcn_wmma_f32_16x16x4_f32(false, a, false, bb,
                                                  (short)0, c, false, false);
      }
      acc[ht] = c;
    }
  }

  // Epilogue: optional ReLU, write C/D-layout accumulators to global.
  float* op = out + ((size_t)b * Nn + p0 + wave * 16) * Dd;
#pragma unroll
  for (int ht = 0; ht < 8; ++ht) {
#pragma unroll
    for (int r = 0; r < 8; ++r) {
      float v = acc[ht][r];
      if (relu) v = fmaxf(v, 0.f);
      op[(size_t)(r + 8 * hi) * Dd + ht * 16 + m] = v;
    }
  }
}

// ---------------------------------------------------------------------------
extern "C" void kernel_launch(void* const* d_in, const int* in_sizes, int n_in,
                              void* d_out, int out_size, void* d_ws, size_t ws_size,
                              hipStream_t stream) {
  const float* x  = (const float*)d_in[0];
  const float* ew = (const float*)d_in[1];
  const float* Wp[3] = {(const float*)d_in[2], (const float*)d_in[4],
                        (const float*)d_in[6]};
  const float* bp[3] = {(const float*)d_in[3], (const float*)d_in[5],
                        (const float*)d_in[7]};
  float* out = (float*)d_out;
  float* ws  = (float*)d_ws;

  const size_t sz = (size_t)Bb * Nn * Dd;   // 2M floats = 8 MB per buffer
  float* hbuf  = ws;
  float* nhbuf = ws + sz;
  float* iobuf = ws + 2 * sz;

  const float* cur = x;
  for (int layer = 0; layer < 3; ++layer) {
    adaptgnn_linear_norm<<<dim3(Bb * Nn), dim3(128), 0, stream>>>(
        cur, Wp[layer], bp[layer], hbuf, nhbuf);
    float* dst = (layer == 2) ? out : iobuf;
    adaptgnn_fused_cos_agg<<<dim3(Bb * (Nn / 64)), dim3(128), 0, stream>>>(
        hbuf, nhbuf, ew, dst, (layer == 2) ? 0 : 1);
    cur = iobuf;
  }
}